// DynamicDirectionalConv_29248727286310
// MI455X (gfx1250) — compile-verified
//
#include <hip/hip_runtime.h>
#include <math.h>

typedef __attribute__((ext_vector_type(2))) float v2f;
typedef __attribute__((ext_vector_type(4))) float v4f;
typedef __attribute__((ext_vector_type(8))) float v8f;

#define WMMA4(a, b, c) \
  __builtin_amdgcn_wmma_f32_16x16x4_f32(false, (a), false, (b), (short)0, (c), false, false)

constexpr int Bn  = 8;
constexpr int Cn  = 256;
constexpr int Hn  = 128;
constexpr int Wn  = 128;
constexpr int MID = 16;
constexpr int HW  = Hn * Wn;
constexpr float FMAXV = 3.402823466e38f;

__device__ __forceinline__ int reflect_idx(int t) {
  // jnp.pad 'reflect' (no edge repeat): -1 -> 1, 128 -> 126
  t = (t < 0) ? -t : t;
  t = (t >= Hn) ? (2 * Hn - 2 - t) : t;
  return t;
}

// ---------------------------------------------------------------------------
// Pass 1: x_low[b, p, m] = sum_c w_reduce[m, c] * x[b, c, p]
// One wave -> 16(mid) x 64(pixel): 4 WMMA D-tiles sharing one A load per
// K-step. Per step: 1x b64 (A) + 8x b32 (x, 1KB useful) + 4 WMMA.
// Output layout: ws[((b*HW + p) * 16) + m]   (mid fastest, 64B per pixel)
// ---------------------------------------------------------------------------
__global__ __launch_bounds__(256) void reduce_gemm_kernel(
    const float* __restrict__ x, const float* __restrict__ w_reduce,
    float* __restrict__ xlow) {
  const int lane = threadIdx.x & 31;
  const int wave = blockIdx.x * 8 + (threadIdx.x >> 5);  // 0..2047
  const int b     = wave >> 8;                           // 64-pixel tiles: 256/b
  const int pbase = (wave & 255) << 6;
  const int half  = lane >> 4;                           // 0: lanes 0-15, 1: 16-31
  const int l16   = lane & 15;
  const int pcol  = pbase + l16;                         // pixel col of tile t=0

  const float* __restrict__ xb   = x + (size_t)b * Cn * HW;
  const float* __restrict__ wrow = w_reduce + l16 * Cn;  // A row M = l16

  v8f acc0 = {}, acc1 = {}, acc2 = {}, acc3 = {};
#pragma unroll 2
  for (int k0 = 0; k0 < Cn; k0 += 4) {
    const int ka = k0 + 2 * half;  // lane-half K pair
    v2f a;
    a.x = wrow[ka];
    a.y = wrow[ka + 1];
    const float* r0 = xb + (size_t)ka * HW + pcol;
    const float* r1 = r0 + HW;
    v2f b0, b1, b2, b3;
    b0.x = r0[0];  b0.y = r1[0];
    b1.x = r0[16]; b1.y = r1[16];
    b2.x = r0[32]; b2.y = r1[32];
    b3.x = r0[48]; b3.y = r1[48];
    acc0 = WMMA4(a, b0, acc0);
    acc1 = WMMA4(a, b1, acc1);
    acc2 = WMMA4(a, b2, acc2);
    acc3 = WMMA4(a, b3, acc3);
  }

  // D layout: lane l, VGPR r -> M = r + 8*half, N = l16. Mids contiguous in ws.
  float* dst = xlow + ((size_t)b * HW + pcol) * MID + 8 * half;
  const v8f accs[4] = {acc0, acc1, acc2, acc3};
#pragma unroll
  for (int t = 0; t < 4; ++t) {
    const v4f* ap = (const v4f*)&accs[t];
    float* d = dst + (size_t)(16 * t) * MID;
    *(v4f*)(d)     = ap[0];
    *(v4f*)(d + 4) = ap[1];
  }
}

// ---------------------------------------------------------------------------
// Pass 2: per-pixel 7x7 directional blur on x_low (reflect pad, separable
// bases generated on the fly), nan_to_num, then expand GEMM 16 -> 256 via
// WMMA, clamp, store to out (B, C, H, W).
// Block: 128 threads = 4 waves; covers one (b, h, 64-wide) row segment.
// ---------------------------------------------------------------------------
#define CELL_STRIDE 20  // 16 mids padded to 20 floats: 16B-aligned, bank-friendly

__global__ __launch_bounds__(128) void conv_expand_kernel(
    const float* __restrict__ xlow, const float* __restrict__ angle,
    const float* __restrict__ w_expand, float* __restrict__ out) {
  __shared__ float lds[7 * 70 * CELL_STRIDE];

  const int blk = blockIdx.x;         // 0..2047 = b(8) * h(128) * tw(2)
  const int tw  = blk & 1;
  const int h   = (blk >> 1) & 127;
  const int b   = blk >> 8;
  const int w0  = tw * 64;

  const int tid = threadIdx.x;

  // --- cooperative halo load: 7 rows x 70 cols x 16 mids, as float4 chunks ---
  for (int t = tid; t < 7 * 70 * 4; t += 128) {
    const int cell = t >> 2;
    const int part = t & 3;
    const int i = cell / 70;
    const int j = cell % 70;
    const int gh = reflect_idx(h - 3 + i);
    const int gw = reflect_idx(w0 - 3 + j);
    const v4f v = *(const v4f*)(xlow + ((size_t)b * HW + gh * Wn + gw) * MID + part * 4);
    *(v4f*)(&lds[cell * CELL_STRIDE + part * 4]) = v;
  }
  __syncthreads();

  const int lane  = tid & 31;
  const int wv    = tid >> 5;       // wave in block: 0..3
  const int l16   = lane & 15;
  const int half  = lane >> 4;
  const int wloc  = wv * 16 + l16;  // local pixel column 0..63
  const int wpix  = w0 + wloc;      // global w

  // --- per-pixel blend weight and separable Gaussian bases ---
  const float ang = angle[(size_t)b * HW + h * Wn + wpix];
  const float cs  = __cosf(ang);
  const float wh  = cs * cs;

  float eh[7], ev[7];
  float sh = 0.f, sv = 0.f;
#pragma unroll
  for (int d = 0; d < 7; ++d) {
    const float dd = (float)(d - 3);
    eh[d] = __expf(-dd * dd * (1.0f / 12.5f));  // sigma_h = 2.5
    ev[d] = __expf(-dd * dd * 0.5f);            // sigma_v = 1.0
    sh += eh[d];
    sv += ev[d];
  }
  const float invS = 1.0f / (sh * sv + 1e-8f);
  const float wA = wh * invS;            // weight of BASE_H
  const float wB = (1.0f - wh) * invS;   // weight of BASE_V = BASE_H^T

  // --- 49-tap conv: lane accumulates (pixel wloc) x (8 mids in WMMA-B order)
  // positions: 2*half + {0,1, 4,5, 8,9, 12,13}
  float acc[8] = {0.f, 0.f, 0.f, 0.f, 0.f, 0.f, 0.f, 0.f};
#pragma unroll
  for (int i = 0; i < 7; ++i) {
#pragma unroll
    for (int j = 0; j < 7; ++j) {
      const float wt = wA * (eh[i] * ev[j]) + wB * (eh[j] * ev[i]);
      const float* cp = &lds[(i * 70 + wloc + j) * CELL_STRIDE + 2 * half];
      const v2f p0 = *(const v2f*)(cp);
      const v2f p1 = *(const v2f*)(cp + 4);
      const v2f p2 = *(const v2f*)(cp + 8);
      const v2f p3 = *(const v2f*)(cp + 12);
      acc[0] = fmaf(wt, p0.x, acc[0]);
      acc[1] = fmaf(wt, p0.y, acc[1]);
      acc[2] = fmaf(wt, p1.x, acc[2]);
      acc[3] = fmaf(wt, p1.y, acc[3]);
      acc[4] = fmaf(wt, p2.x, acc[4]);
      acc[5] = fmaf(wt, p2.y, acc[5]);
      acc[6] = fmaf(wt, p3.x, acc[6]);
      acc[7] = fmaf(wt, p3.y, acc[7]);
    }
  }

  // nan_to_num on out_low
#pragma unroll
  for (int q = 0; q < 8; ++q) {
    float a = acc[q];
    a = (a != a) ? 0.0f : fminf(fmaxf(a, -FMAXV), FMAXV);
    acc[q] = a;
  }

  // acc already matches WMMA B layout: step s uses K pair (4s + 2*half, +1)
  v2f bmat[4];
#pragma unroll
  for (int s = 0; s < 4; ++s) {
    bmat[s].x = acc[2 * s];
    bmat[s].y = acc[2 * s + 1];
  }

  // --- expand GEMM: out[c, pix] = sum_m w_expand[c, m] * out_low[m, pix] ---
  const size_t outbase = (size_t)b * Cn * HW + (size_t)h * Wn + wpix;
#pragma unroll 4
  for (int ct = 0; ct < 16; ++ct) {
    const float* __restrict__ arow = w_expand + (ct * 16 + l16) * MID;  // A row M
    v8f d = {};
#pragma unroll
    for (int s = 0; s < 4; ++s) {
      const int k = 4 * s + 2 * half;
      v2f a;
      a.x = arow[k];
      a.y = arow[k + 1];
      d = WMMA4(a, bmat[s], d);
    }
    // D: lane l, VGPR r -> channel ct*16 + r + 8*half, pixel = l16 (our wpix)
    const float* dv = (const float*)&d;
    float* op = out + outbase + (size_t)(ct * 16 + 8 * half) * HW;
#pragma unroll
    for (int r = 0; r < 8; ++r) {
      float v = dv[r];
      v = fminf(fmaxf(v, -FMAXV), FMAXV);  // final nan_to_num (values finite)
      op[(size_t)r * HW] = v;
    }
  }
}

// ---------------------------------------------------------------------------
extern "C" void kernel_launch(void* const* d_in, const int* in_sizes, int n_in,
                              void* d_out, int out_size, void* d_ws, size_t ws_size,
                              hipStream_t stream) {
  (void)in_sizes; (void)n_in; (void)out_size; (void)ws_size;
  const float* x        = (const float*)d_in[0];
  const float* angle    = (const float*)d_in[1];
  const float* w_reduce = (const float*)d_in[2];
  const float* w_expand = (const float*)d_in[3];
  float* out  = (float*)d_out;
  float* xlow = (float*)d_ws;  // 8 * 16384 * 16 floats = 8 MB (L2-resident)

  // Pass 1: 2048 waves (16 mids x 64 pixels each), 8 per block
  hipLaunchKernelGGL(reduce_gemm_kernel, dim3(256), dim3(256), 0, stream,
                     x, w_reduce, xlow);
  // Pass 2: b(8) * h(128) * tw(2) = 2048 blocks of 128 threads
  hipLaunchKernelGGL(conv_expand_kernel, dim3(2048), dim3(128), 0, stream,
                     xlow, angle, w_expand, out);
}